// RNN_19576460936017
// MI455X (gfx1250) — compile-verified
//
#include <hip/hip_runtime.h>
#include <cmath>
#include <cstddef>

typedef __attribute__((ext_vector_type(2))) float v2f;
typedef __attribute__((ext_vector_type(8))) float v8f;

#define SEQ    28
#define NBATCH 8192
#define NIN    28
#define HID    128
#define NOUT   10
#define GATES  512     // 4*HID
#define APAD   132     // LDS row stride (mod 64 = 4 -> conflict-free column reads)
#define MTILES 4       // 16-row M tiles per wave -> 64 batch rows per block
#define MROWS  (MTILES * 16)

__device__ __forceinline__ float sigmoidf_(float x) {
    return 1.0f / (1.0f + expf(-x));
}

// ---------------------------------------------------------------------------
// Fused LSTM step. Block = 64 batch rows x all 512 gate columns, 8 waves.
// Wave w owns hidden columns [16w,16w+16) of each gate and 4 M tiles, so each
// B-operand register pair feeds 4 WMMAs (4x less weight traffic from L2).
// gates = A1 @ Wt1 + h_prev @ Wt2 + bias, then i/f/g/o nonlinearity in-regs.
// ---------------------------------------------------------------------------
__global__ __launch_bounds__(256) void lstm_step_kernel(
    const float* __restrict__ A1, int K1,
    const float* __restrict__ Wt1,    // [K1][512]  (pre-transposed W_ih)
    const float* __restrict__ A2,     // [B][128]   (h_prev)
    const float* __restrict__ Wt2,    // [128][512] (pre-transposed W_hh)
    const float* __restrict__ bias,   // [512]      (b_ih + b_hh)
    float* __restrict__ c,            // [B][128] in/out
    float* __restrict__ h_out)        // [B][128]
{
    __shared__ float sA1[MROWS * APAD];
    __shared__ float sA2[MROWS * APAD];

    const int tid      = threadIdx.x;
    const int wave     = tid >> 5;
    const int lane     = tid & 31;
    const int lane16   = lane & 15;
    const int khalf    = (lane >> 4) << 1;   // 0 for lanes 0-15, 2 for lanes 16-31
    const int row_base = blockIdx.x * MROWS;

    // Cooperative, coalesced float4 staging of the two A tiles into padded LDS.
    {
        const int K1v = K1 >> 2;  // K1 in {28,128}, both multiples of 4
        for (int idx = tid; idx < MROWS * K1v; idx += 256) {
            int r = idx / K1v, c4 = idx - r * K1v;
            float4 v = ((const float4*)(A1 + (size_t)(row_base + r) * K1))[c4];
            *(float4*)&sA1[r * APAD + c4 * 4] = v;
        }
    }
    for (int idx = tid; idx < MROWS * (HID / 4); idx += 256) {
        int r = idx >> 5, c4 = idx & 31;
        float4 v = ((const float4*)(A2 + (size_t)(row_base + r) * HID))[c4];
        *(float4*)&sA2[r * APAD + c4 * 4] = v;
    }
    __syncthreads();

    const int n_base = wave * 16;  // column inside each 128-wide gate block

    // acc[m][g]: M tile m, gate g (i,f,g,o). Init with folded bias.
    v8f acc[MTILES][4];
#pragma unroll
    for (int g = 0; g < 4; ++g) {
        float bv = bias[g * HID + n_base + lane16];
#pragma unroll
        for (int m = 0; m < MTILES; ++m)
#pragma unroll
            for (int r = 0; r < 8; ++r) acc[m][g][r] = bv;
    }

    // ---- part 1: input contribution (K = K1) ----
    for (int kk = 0; kk < K1; kk += 4) {
        v2f a[MTILES];
#pragma unroll
        for (int m = 0; m < MTILES; ++m) {
            const float* sp = sA1 + (m * 16 + lane16) * APAD + kk + khalf;
            a[m].x = sp[0];
            a[m].y = sp[1];
        }
        const float* w0 = Wt1 + (size_t)(kk + khalf) * GATES + n_base + lane16;
#pragma unroll
        for (int g = 0; g < 4; ++g) {
            v2f b;
            b.x = w0[g * HID];
            b.y = w0[g * HID + GATES];
#pragma unroll
            for (int m = 0; m < MTILES; ++m)
                acc[m][g] = __builtin_amdgcn_wmma_f32_16x16x4_f32(
                    false, a[m], false, b, (short)0, acc[m][g], false, false);
        }
    }
    // ---- part 2: recurrent contribution (K = 128) ----
    for (int kk = 0; kk < HID; kk += 4) {
        v2f a[MTILES];
#pragma unroll
        for (int m = 0; m < MTILES; ++m) {
            const float* sp = sA2 + (m * 16 + lane16) * APAD + kk + khalf;
            a[m].x = sp[0];
            a[m].y = sp[1];
        }
        const float* w0 = Wt2 + (size_t)(kk + khalf) * GATES + n_base + lane16;
#pragma unroll
        for (int g = 0; g < 4; ++g) {
            v2f b;
            b.x = w0[g * HID];
            b.y = w0[g * HID + GATES];
#pragma unroll
            for (int m = 0; m < MTILES; ++m)
                acc[m][g] = __builtin_amdgcn_wmma_f32_16x16x4_f32(
                    false, a[m], false, b, (short)0, acc[m][g], false, false);
        }
    }

    // ---- fused LSTM pointwise, wave-local (C layout: VGPR r -> M=r / r+8) ----
    const int col = n_base + lane16;
#pragma unroll
    for (int m = 0; m < MTILES; ++m) {
#pragma unroll
        for (int r = 0; r < 8; ++r) {
            int row = row_base + m * 16 + r + ((lane >> 4) << 3);
            size_t off = (size_t)row * HID + col;
            float iv = sigmoidf_(acc[m][0][r]);
            float fv = sigmoidf_(acc[m][1][r]);
            float gv = tanhf(acc[m][2][r]);
            float ov = sigmoidf_(acc[m][3][r]);
            float cn = fv * c[off] + iv * gv;
            c[off]     = cn;
            h_out[off] = ov * tanhf(cn);
        }
    }
}

// W [512][K] row-major  ->  Wt [K][512] row-major (coalesced B-operand reads)
__global__ void transpose_kernel(const float* __restrict__ in,
                                 float* __restrict__ out, int K) {
    int idx = blockIdx.x * blockDim.x + threadIdx.x;
    if (idx < GATES * K) {
        int n = idx / K, k = idx - n * K;
        out[(size_t)k * GATES + n] = in[idx];
    }
}

__global__ void bias_fold_kernel(const float* __restrict__ b_ih0,
                                 const float* __restrict__ b_hh0,
                                 const float* __restrict__ b_ih1,
                                 const float* __restrict__ b_hh1,
                                 float* __restrict__ bias0,
                                 float* __restrict__ bias1) {
    int i = blockIdx.x * blockDim.x + threadIdx.x;
    if (i < GATES) {
        bias0[i] = b_ih0[i] + b_hh0[i];
        bias1[i] = b_ih1[i] + b_hh1[i];
    }
}

__global__ void zero_kernel(float* __restrict__ p, size_t n) {
    size_t i = (size_t)blockIdx.x * blockDim.x + threadIdx.x;
    if (i < n) p[i] = 0.0f;
}

// pred[B,10] = h2_last @ W_lin^T + b_lin
__global__ __launch_bounds__(256) void linear_kernel(
    const float* __restrict__ h, const float* __restrict__ W,
    const float* __restrict__ b, float* __restrict__ out) {
    int idx = blockIdx.x * blockDim.x + threadIdx.x;  // B*16 threads
    int bb = idx >> 4, o = idx & 15;
    if (bb < NBATCH && o < NOUT) {
        float s = b[o];
        const float* hr = h + (size_t)bb * HID;
        const float* wr = W + o * HID;
#pragma unroll 16
        for (int k = 0; k < HID; ++k) s += hr[k] * wr[k];
        out[(size_t)bb * NOUT + o] = s;
    }
}

extern "C" void kernel_launch(void* const* d_in, const int* in_sizes, int n_in,
                              void* d_out, int out_size, void* d_ws, size_t ws_size,
                              hipStream_t stream) {
    const float* X     = (const float*)d_in[0];
    const float* W_ih0 = (const float*)d_in[1];
    const float* W_hh0 = (const float*)d_in[2];
    const float* b_ih0 = (const float*)d_in[3];
    const float* b_hh0 = (const float*)d_in[4];
    const float* W_ih1 = (const float*)d_in[5];
    const float* W_hh1 = (const float*)d_in[6];
    const float* b_ih1 = (const float*)d_in[7];
    const float* b_hh1 = (const float*)d_in[8];
    const float* W_lin = (const float*)d_in[9];
    const float* b_lin = (const float*)d_in[10];
    float* out = (float*)d_out;

    float* ws = (float*)d_ws;
    size_t off = 0;
    float* Wt_ih0 = ws + off; off += (size_t)NIN * GATES;   // [28][512]
    float* Wt_hh0 = ws + off; off += (size_t)HID * GATES;   // [128][512]
    float* Wt_ih1 = ws + off; off += (size_t)HID * GATES;
    float* Wt_hh1 = ws + off; off += (size_t)HID * GATES;
    float* bias0  = ws + off; off += GATES;
    float* bias1  = ws + off; off += GATES;
    const size_t HB = (size_t)NBATCH * HID;                 // 1,048,576 floats
    float* state  = ws + off;                               // 6 contiguous h/c buffers
    float* c0  = state + 0 * HB;
    float* c1  = state + 1 * HB;
    float* h1a = state + 2 * HB;
    float* h1b = state + 3 * HB;
    float* h2a = state + 4 * HB;
    float* h2b = state + 5 * HB;

    // ---- prep: transposes, bias fold, zero state ----
    transpose_kernel<<<(GATES * NIN + 255) / 256, 256, 0, stream>>>(W_ih0, Wt_ih0, NIN);
    transpose_kernel<<<(GATES * HID + 255) / 256, 256, 0, stream>>>(W_hh0, Wt_hh0, HID);
    transpose_kernel<<<(GATES * HID + 255) / 256, 256, 0, stream>>>(W_ih1, Wt_ih1, HID);
    transpose_kernel<<<(GATES * HID + 255) / 256, 256, 0, stream>>>(W_hh1, Wt_hh1, HID);
    bias_fold_kernel<<<2, 256, 0, stream>>>(b_ih0, b_hh0, b_ih1, b_hh1, bias0, bias1);
    {
        size_t n = 6 * HB;
        zero_kernel<<<(unsigned)((n + 255) / 256), 256, 0, stream>>>(state, n);
    }

    // ---- 28 timesteps, two fused layer steps each ----
    float* h1_prev = h1a; float* h1_cur = h1b;
    float* h2_prev = h2a; float* h2_cur = h2b;
    const int nblk = NBATCH / MROWS;  // 128

    for (int t = 0; t < SEQ; ++t) {
        const float* xt = X + (size_t)t * NBATCH * NIN;
        lstm_step_kernel<<<nblk, 256, 0, stream>>>(
            xt, NIN, Wt_ih0, h1_prev, Wt_hh0, bias0, c0, h1_cur);
        lstm_step_kernel<<<nblk, 256, 0, stream>>>(
            h1_cur, HID, Wt_ih1, h2_prev, Wt_hh1, bias1, c1, h2_cur);
        float* tmp;
        tmp = h1_prev; h1_prev = h1_cur; h1_cur = tmp;
        tmp = h2_prev; h2_prev = h2_cur; h2_cur = tmp;
    }

    // after final swap, last h2 lives in h2_prev
    linear_kernel<<<(NBATCH * 16 + 255) / 256, 256, 0, stream>>>(
        h2_prev, W_lin, b_lin, out);
}